// DeformableAttention2D_63101659513428
// MI455X (gfx1250) — compile-verified
//
#include <hip/hip_runtime.h>
#include <math.h>

// ---------------------------------------------------------------------------
// Deformable attention forward for MI455X (gfx1250), fp32 throughout.
// GEMMs run on V_WMMA_F32_16X16X4_F32 (wave32 matrix pipe).
// ---------------------------------------------------------------------------

#define DMODEL 256
#define NHEADS 8
#define DHEAD  32
#define QTOT   13294
#define BATCH  2
#define MTOT   (BATCH * QTOT)   // 26588
#define KDIM   256

typedef __attribute__((ext_vector_type(2))) float v2f;
typedef __attribute__((ext_vector_type(8))) float v8f;

__device__ __forceinline__ void q_to_level(int q, int& lvl, int& local,
                                           int& H, int& W) {
    if (q < 10000)      { lvl = 0; local = q;         H = 100; W = 100; }
    else if (q < 12500) { lvl = 1; local = q - 10000; H = 50;  W = 50;  }
    else if (q < 13125) { lvl = 2; local = q - 12500; H = 25;  W = 25;  }
    else                { lvl = 3; local = q - 13125; H = 13;  W = 13;  }
}

// ---------------------------------------------------------------------------
// 1) Flatten feats to (B,Q,D) and build query = value_in + sine-pos.
//    pos(d, level, h, w): DETR sine embedding on 100x100 grid, nearest-
//    resampled via yi = h*100//H, xi = w*100//W (computed analytically).
// ---------------------------------------------------------------------------
__global__ __launch_bounds__(256)
void prep_kernel(const float* __restrict__ f0, const float* __restrict__ f1,
                 const float* __restrict__ f2, const float* __restrict__ f3,
                 float* __restrict__ value_in, float* __restrict__ query) {
    const int q = blockIdx.x, b = blockIdx.y, d = threadIdx.x;
    int lvl, local, H, W;
    q_to_level(q, lvl, local, H, W);
    const int h = local / W, w = local - h * W;
    const float* f = (lvl == 0) ? f0 : (lvl == 1) ? f1 : (lvl == 2) ? f2 : f3;
    const float val = f[(((size_t)b * DMODEL + d) * H + h) * W + w];

    const int yi = (h * 100) / H;
    const int xi = (w * 100) / W;
    const int cc = d & 127;
    float coord = (d < 128) ? (yi + 0.5f) : (xi + 0.5f);
    coord *= 0.06283185307179587f;                 // 2*pi/100
    const int k = cc >> 1;
    const float inv_dimt = __expf(-0.14391156831212787f * (float)k); // 10000^{-k/64}
    const float ang = coord * inv_dimt;
    const float pos = (cc & 1) ? __cosf(ang) : __sinf(ang);

    const size_t idx = ((size_t)(b * QTOT + q)) * DMODEL + d;
    value_in[idx] = val;
    query[idx]    = val + pos;
}

// ---------------------------------------------------------------------------
// 2) GEMM: out[m,n] = sum_k A[m,k] * Wt[n,k] + bias[n]  (+ residual, scatter)
//    Block = 8 waves; block tile 16(M) x 128(N); wave tile 16x16.
//    K-loop: 64x v_wmma_f32_16x16x4_f32.
//    A frag  (ISA 16x4 layout): lane<16 -> A[m][k0..k0+1], lane>=16 -> k0+2..3
//    B frag  (K striped across VGPRs, N across lanes): Wt row is contiguous.
//    C/D     : element i of v8f -> M = (lane>=16?8:0)+i, N = lane&15.
// ---------------------------------------------------------------------------
template <bool SCATTER>
__global__ __launch_bounds__(256)
void gemm_wmma(const float* __restrict__ A, const float* __restrict__ Wt,
               const float* __restrict__ bias, const float* __restrict__ residual,
               float* __restrict__ out, int Ntot) {
    __shared__ float a_lds[16 * 260];   // stride 260: 16B-aligned rows, conflict-free
    const int t  = threadIdx.x;
    const int m0 = blockIdx.x * 16;

    // cooperatively stage the 16x256 A tile
    {
        const int row = t >> 4, tc = t & 15;
        const int m = m0 + row;
        const float* arow = A + (size_t)m * KDIM;
#pragma unroll
        for (int j = 0; j < 4; ++j) {
            const int col = tc * 4 + j * 64;
            float4 v4 = make_float4(0.f, 0.f, 0.f, 0.f);
            if (m < MTOT) v4 = *(const float4*)(arow + col);
            *(float4*)(&a_lds[row * 260 + col]) = v4;
        }
    }
    __syncthreads();

    const int lane = t & 31, wave = t >> 5;
    const int half = lane >> 4, l16 = lane & 15;
    const int n = blockIdx.y * 128 + wave * 16 + l16;
    const float* wrow = Wt + (size_t)n * KDIM;

    v8f acc = {};
    for (int k0 = 0; k0 < KDIM; k0 += 4) {
        const int kb = k0 + half * 2;
        v2f a = *(const v2f*)(&a_lds[l16 * 260 + kb]);
        v2f bf = *(const v2f*)(wrow + kb);
        acc = __builtin_amdgcn_wmma_f32_16x16x4_f32(
            false, a, false, bf, (short)0, acc, false, false);
    }

    const float bv = bias[n];
#pragma unroll
    for (int i = 0; i < 8; ++i) {
        const int m = m0 + half * 8 + i;
        if (m < MTOT) {
            float val = acc[i] + bv;
            if (!SCATTER) {
                out[(size_t)m * Ntot + n] = val;
            } else {
                val += residual[(size_t)m * DMODEL + n];
                const int b = m / QTOT, q = m - b * QTOT;
                int lvl, local, H, W;
                q_to_level(q, lvl, local, H, W);
                const int HW = H * W;
                const int obase = (lvl == 0) ? 0 : (lvl == 1) ? 5120000
                                  : (lvl == 2) ? 6400000 : 6720000;
                out[(size_t)obase + ((size_t)(b * DMODEL + n)) * HW + local] = val;
            }
        }
    }
}

// ---------------------------------------------------------------------------
// 3) Per-(b,q): softmax over 16 logits per head, 4x4 bilinear gathers from v,
//    weighted sum -> out_attn (B,Q,D) with d = head*32 + c.
// ---------------------------------------------------------------------------
__global__ __launch_bounds__(256)
void sample_kernel(const float* __restrict__ v, const float* __restrict__ off_raw,
                   const float* __restrict__ attn_raw, float* __restrict__ out_attn) {
    const int q = blockIdx.x, b = blockIdx.y;
    const int t = threadIdx.x;
    const int head = t >> 5, c = t & 31;
    __shared__ float s_off[256];
    __shared__ float s_attn[128];
    const size_t row = (size_t)(b * QTOT + q);
    s_off[t] = off_raw[row * 256 + t];
    if (t < 128) s_attn[t] = attn_raw[row * 128 + t];
    __syncthreads();

    float mx = -1e30f;
#pragma unroll
    for (int j = 0; j < 16; ++j) mx = fmaxf(mx, s_attn[head * 16 + j]);
    float den = 0.f;
#pragma unroll
    for (int j = 0; j < 16; ++j) den += __expf(s_attn[head * 16 + j] - mx);
    const float rden = 1.f / den;

    int lvq, localq, Hq, Wq;
    q_to_level(q, lvq, localq, Hq, Wq);
    const int hq = localq / Wq, wq = localq - hq * Wq;
    const float rx = (wq + 0.5f) / Wq, ry = (hq + 0.5f) / Hq;

    const float* vb = v + (size_t)b * QTOT * DMODEL + head * DHEAD + c;
    float acc = 0.f;
#pragma unroll
    for (int l = 0; l < 4; ++l) {
        const int Hl = (l == 0) ? 100 : (l == 1) ? 50 : (l == 2) ? 25 : 13;
        const int Wl = Hl;
        const int qoffl = (l == 0) ? 0 : (l == 1) ? 10000 : (l == 2) ? 12500 : 13125;
#pragma unroll
        for (int p = 0; p < 4; ++p) {
            const float ox = s_off[((head * 4 + l) * 4 + p) * 2 + 0];
            const float oy = s_off[((head * 4 + l) * 4 + p) * 2 + 1];
            const float aw = __expf(s_attn[head * 16 + l * 4 + p] - mx) * rden;
            // loc = ref + off/(W,H); x = loc_x*W - 0.5 ; grid_sample bilinear, zero pad
            const float x = rx * Wl + ox - 0.5f;
            const float y = ry * Hl + oy - 0.5f;
            const float fx0 = floorf(x), fy0 = floorf(y);
            const float wx = x - fx0, wy = y - fy0;
            const int x0 = (int)fx0, y0 = (int)fy0;
            const int x1 = x0 + 1, y1 = y0 + 1;
            const bool xin0 = (x0 >= 0) && (x0 < Wl), xin1 = (x1 >= 0) && (x1 < Wl);
            const bool yin0 = (y0 >= 0) && (y0 < Hl), yin1 = (y1 >= 0) && (y1 < Hl);
            float v00 = 0.f, v01 = 0.f, v10 = 0.f, v11 = 0.f;
            if (xin0 && yin0) v00 = vb[(size_t)(qoffl + y0 * Wl + x0) * DMODEL];
            if (xin1 && yin0) v01 = vb[(size_t)(qoffl + y0 * Wl + x1) * DMODEL];
            if (xin0 && yin1) v10 = vb[(size_t)(qoffl + y1 * Wl + x0) * DMODEL];
            if (xin1 && yin1) v11 = vb[(size_t)(qoffl + y1 * Wl + x1) * DMODEL];
            const float bil = v00 * (1.f - wx) * (1.f - wy) + v01 * wx * (1.f - wy)
                            + v10 * (1.f - wx) * wy + v11 * wx * wy;
            acc += aw * bil;
        }
    }
    out_attn[row * DMODEL + head * DHEAD + c] = acc;
}

// ---------------------------------------------------------------------------
// launcher
// ---------------------------------------------------------------------------
extern "C" void kernel_launch(void* const* d_in, const int* in_sizes, int n_in,
                              void* d_out, int out_size, void* d_ws, size_t ws_size,
                              hipStream_t stream) {
    (void)in_sizes; (void)n_in; (void)out_size; (void)ws_size;
    const float* f0      = (const float*)d_in[0];
    const float* f1      = (const float*)d_in[1];
    const float* f2      = (const float*)d_in[2];
    const float* f3      = (const float*)d_in[3];
    const float* w_value = (const float*)d_in[4];
    const float* b_value = (const float*)d_in[5];
    const float* w_off   = (const float*)d_in[6];
    const float* b_off   = (const float*)d_in[7];
    const float* w_attn  = (const float*)d_in[8];
    const float* b_attn  = (const float*)d_in[9];
    const float* w_out   = (const float*)d_in[10];
    const float* b_out   = (const float*)d_in[11];

    float* ws = (float*)d_ws;
    const size_t R = (size_t)MTOT * DMODEL;   // 6,806,528 floats per region
    float* value_in = ws;                     // (B,Q,256)  kept for residual
    float* query    = ws + R;                 // (B,Q,256)  reused as out_attn
    float* vproj    = ws + 2 * R;             // (B,Q,256)
    float* off_raw  = ws + 3 * R;             // (B,Q,256)
    float* attn_raw = ws + 4 * R;             // (B,Q,128)

    const dim3 gq(QTOT, BATCH);
    const dim3 gg((MTOT + 15) / 16, 2);       // N=256 -> 2 column groups of 128
    const dim3 ga((MTOT + 15) / 16, 1);       // N=128 -> 1 column group

    prep_kernel<<<gq, 256, 0, stream>>>(f0, f1, f2, f3, value_in, query);
    gemm_wmma<false><<<gg, 256, 0, stream>>>(value_in, w_value, b_value, nullptr, vproj, 256);
    gemm_wmma<false><<<gg, 256, 0, stream>>>(query, w_off, b_off, nullptr, off_raw, 256);
    gemm_wmma<false><<<ga, 256, 0, stream>>>(query, w_attn, b_attn, nullptr, attn_raw, 128);
    sample_kernel<<<gq, 256, 0, stream>>>(vproj, off_raw, attn_raw, /*out_attn=*/query);
    gemm_wmma<true><<<gg, 256, 0, stream>>>(query, w_out, b_out, value_in, (float*)d_out, 256);
}